// ResidualBlock_4612794876592
// MI455X (gfx1250) — compile-verified
//
#include <hip/hip_runtime.h>
#include <cmath>

#define D_DIM 128
#define LRELU_SLOPE 0.01f

typedef float v2f __attribute__((ext_vector_type(2)));
typedef float v8f __attribute__((ext_vector_type(8)));

// ---------------------------------------------------------------------------
// 1) zero the BN stats accumulators (256 floats: colsum + colsumsq)
// ---------------------------------------------------------------------------
__global__ void zero_stats_kernel(float* __restrict__ stats) {
  stats[threadIdx.x] = 0.0f;
}

// ---------------------------------------------------------------------------
// 2) per-column partial sums of x and x^2 (coalesced row reads, L2 atomics)
// ---------------------------------------------------------------------------
__global__ void bn_partial_kernel(const float* __restrict__ x,
                                  float* __restrict__ stats, int n) {
  int c = threadIdx.x;  // 0..127
  float s = 0.0f, s2 = 0.0f;
  for (int r = blockIdx.x; r < n; r += gridDim.x) {
    float v = x[(size_t)r * D_DIM + c];
    s += v;
    s2 += v * v;
  }
  atomicAdd(&stats[c], s);
  atomicAdd(&stats[D_DIM + c], s2);
}

// ---------------------------------------------------------------------------
// 3) fold stats -> per-column scale/shift for BN
// ---------------------------------------------------------------------------
__global__ void bn_finalize_kernel(const float* __restrict__ stats,
                                   const float* __restrict__ gamma,
                                   const float* __restrict__ beta,
                                   float* __restrict__ scale_shift, int n) {
  int c = threadIdx.x;  // 0..127
  float invn = 1.0f / (float)n;
  float mean = stats[c] * invn;
  float var  = stats[D_DIM + c] * invn - mean * mean;  // biased variance
  float sc = gamma[c] * rsqrtf(var + 1e-5f);
  scale_shift[c]         = sc;
  scale_shift[D_DIM + c] = beta[c] - mean * sc;
}

// ---------------------------------------------------------------------------
// 4) transpose a 128x128 weight matrix to column-major so the WMMA B-fragment
//    pair (K, K+1) is a contiguous 8-byte load (global_load_b64)
// ---------------------------------------------------------------------------
__global__ void transpose_w_kernel(const float* __restrict__ W,
                                   float* __restrict__ Wt) {
  int t = blockIdx.x * blockDim.x + threadIdx.x;  // 16384 threads
  int k = t >> 7;
  int c = t & (D_DIM - 1);
  Wt[(size_t)c * D_DIM + k] = W[(size_t)k * D_DIM + c];
}

// ---------------------------------------------------------------------------
// 5) h = LeakyReLU(BN(x));  agg = (1 + eps_gin) * h   (float4 streaming)
// ---------------------------------------------------------------------------
__global__ void bn_apply_kernel(const float* __restrict__ x,
                                const float* __restrict__ scale_shift,
                                const float* __restrict__ epsp,
                                float* __restrict__ h,
                                float* __restrict__ agg, int n) {
  int i = blockIdx.x * blockDim.x + threadIdx.x;
  int total = n * (D_DIM / 4);
  if (i >= total) return;
  int c4 = i & (D_DIM / 4 - 1);
  float4 xv = ((const float4*)x)[i];
  float4 sc = ((const float4*)scale_shift)[c4];
  float4 sh = ((const float4*)(scale_shift + D_DIM))[c4];
  float4 hv;
  hv.x = fmaf(xv.x, sc.x, sh.x); hv.x = hv.x >= 0.0f ? hv.x : LRELU_SLOPE * hv.x;
  hv.y = fmaf(xv.y, sc.y, sh.y); hv.y = hv.y >= 0.0f ? hv.y : LRELU_SLOPE * hv.y;
  hv.z = fmaf(xv.z, sc.z, sh.z); hv.z = hv.z >= 0.0f ? hv.z : LRELU_SLOPE * hv.z;
  hv.w = fmaf(xv.w, sc.w, sh.w); hv.w = hv.w >= 0.0f ? hv.w : LRELU_SLOPE * hv.w;
  ((float4*)h)[i] = hv;
  float g = 1.0f + epsp[0];
  float4 av; av.x = g * hv.x; av.y = g * hv.y; av.z = g * hv.z; av.w = g * hv.w;
  ((float4*)agg)[i] = av;
}

// ---------------------------------------------------------------------------
// 6) edge kernel: w = ELU(LeakyReLU(a*We1+be1)@We2+be2); agg[dst] += w*h[src]
//    4 threads per edge, 32 columns each; h/agg are L2-resident (25.6 MB each)
// ---------------------------------------------------------------------------
__global__ void edge_scatter_kernel(const float* __restrict__ h,
                                    const int* __restrict__ ei,
                                    const float* __restrict__ ea,
                                    const float* __restrict__ We1,
                                    const float* __restrict__ be1,
                                    const float* __restrict__ We2,
                                    const float* __restrict__ be2,
                                    float* __restrict__ agg, int ne) {
  int tid = blockIdx.x * blockDim.x + threadIdx.x;
  int e = tid >> 2;
  if (e >= ne) return;
  int g = tid & 3;  // column group: 32 cols

  // tiny edge MLP (redundant across the 4 threads of this edge; all L0 hits)
  float a = ea[e];
  float w = be2[0];
#pragma unroll
  for (int j = 0; j < 8; ++j) {
    float t = fmaf(a, We1[j], be1[j]);
    t = t >= 0.0f ? t : LRELU_SLOPE * t;
    w = fmaf(t, We2[j], w);
  }
  w = w > 0.0f ? w : expm1f(w);  // ELU(alpha=1)

  int src = ei[e];
  int dst = ei[ne + e];
  const float4* hrow = (const float4*)(h + (size_t)src * D_DIM + g * 32);
  float* arow = agg + (size_t)dst * D_DIM + g * 32;
#pragma unroll
  for (int q = 0; q < 8; ++q) {
    float4 v = hrow[q];
    atomicAdd(arow + q * 4 + 0, w * v.x);
    atomicAdd(arow + q * 4 + 1, w * v.y);
    atomicAdd(arow + q * 4 + 2, w * v.z);
    atomicAdd(arow + q * 4 + 3, w * v.w);
  }
}

// ---------------------------------------------------------------------------
// 7/8) WMMA f32 GEMM: Out = [LeakyReLU](A @ W + bias) [+ resid]
//      A:[n,128] row-major, Wt:[128,128] COLUMN-major. Block = 256 thr
//      (8 waves); block owns 16 rows, wave w owns 16 columns.
//      V_WMMA_F32_16X16X4_F32, K stepped by 4; B pair is one b64 load.
//      Epilogue bounces through LDS for fully coalesced b128 stores.
// ---------------------------------------------------------------------------
__global__ void node_mlp_gemm_kernel(const float* A,  // may alias Out
                                     const float* __restrict__ Wt,
                                     const float* __restrict__ bias,
                                     const float* __restrict__ resid,
                                     float* Out,
                                     int n, int do_lrelu) {
  __shared__ float Atile[16 * D_DIM];  // 8 KB

  int row_base = blockIdx.x * 16;
  int t = threadIdx.x;

  // stage 16x128 A tile to LDS (float4, coalesced); zero-pad tail rows
  for (int p = t; p < 16 * (D_DIM / 4); p += 256) {
    int row = p >> 5;
    int gr = row_base + row;
    float4 v;
    if (gr < n) {
      v = ((const float4*)A)[(size_t)gr * (D_DIM / 4) + (p & 31)];
    } else {
      v.x = v.y = v.z = v.w = 0.0f;
    }
    ((float4*)Atile)[p] = v;
  }
  __syncthreads();

  int wave = t >> 5;
  int lane = t & 31;
  int lo = lane & 15;
  int hi = lane >> 4;          // 0: K,K+1 half; 1: K+2,K+3 half
  int col = wave * 16 + lo;    // output column this lane owns
  const float* wcol = Wt + (size_t)col * D_DIM;  // contiguous K for this col

  v8f acc = {};
#pragma unroll
  for (int k = 0; k < D_DIM; k += 4) {
    v2f a, b;
    a.x = Atile[lo * D_DIM + k + 2 * hi];
    a.y = Atile[lo * D_DIM + k + 2 * hi + 1];
    b = *(const v2f*)(wcol + k + 2 * hi);  // 8B aligned: k+2*hi is even
    acc = __builtin_amdgcn_wmma_f32_16x16x4_f32(
        /*neg_a=*/false, a, /*neg_b=*/false, b,
        /*c_mod=*/(short)0, acc, /*reuse_a=*/false, /*reuse_b=*/false);
  }

  // all waves done reading the A tile -> reuse LDS for the output tile
  __syncthreads();
#pragma unroll
  for (int r = 0; r < 8; ++r) {
    // C/D layout: VGPR r -> rows (hi*8 + r), column `col`
    Atile[(hi * 8 + r) * D_DIM + col] = acc[r];
  }
  __syncthreads();

  // coalesced float4 epilogue: bias + activation + residual + b128 store
  for (int p = t; p < 16 * (D_DIM / 4); p += 256) {
    int row = p >> 5;
    int gr = row_base + row;
    if (gr >= n) continue;
    int c4 = p & 31;
    float4 v = ((const float4*)Atile)[p];
    float4 bv = ((const float4*)bias)[c4];
    v.x += bv.x; v.y += bv.y; v.z += bv.z; v.w += bv.w;
    if (do_lrelu) {
      v.x = v.x >= 0.0f ? v.x : LRELU_SLOPE * v.x;
      v.y = v.y >= 0.0f ? v.y : LRELU_SLOPE * v.y;
      v.z = v.z >= 0.0f ? v.z : LRELU_SLOPE * v.z;
      v.w = v.w >= 0.0f ? v.w : LRELU_SLOPE * v.w;
    }
    if (resid) {
      float4 rv = ((const float4*)resid)[(size_t)gr * (D_DIM / 4) + c4];
      v.x += rv.x; v.y += rv.y; v.z += rv.z; v.w += rv.w;
    }
    ((float4*)Out)[(size_t)gr * (D_DIM / 4) + c4] = v;
  }
}

// ---------------------------------------------------------------------------
// launch
// ---------------------------------------------------------------------------
extern "C" void kernel_launch(void* const* d_in, const int* in_sizes, int n_in,
                              void* d_out, int out_size, void* d_ws, size_t ws_size,
                              hipStream_t stream) {
  const float* x     = (const float*)d_in[0];
  const int*   ei    = (const int*)d_in[1];   // edge_index [2, E]
  const float* ea    = (const float*)d_in[2]; // edge_attr  [E, 1]
  const float* gamma = (const float*)d_in[3];
  const float* beta  = (const float*)d_in[4];
  const float* epsp  = (const float*)d_in[5];
  const float* We1   = (const float*)d_in[6];
  const float* be1   = (const float*)d_in[7];
  const float* We2   = (const float*)d_in[8];
  const float* be2   = (const float*)d_in[9];
  const float* W1    = (const float*)d_in[10];
  const float* b1    = (const float*)d_in[11];
  const float* W2    = (const float*)d_in[12];
  const float* b2    = (const float*)d_in[13];

  int n  = in_sizes[0] / D_DIM;  // 50000
  int ne = in_sizes[2];          // E = 1600000 (edge_attr has E*1 elements)

  float* ws    = (float*)d_ws;
  float* stats = ws;        // 256 floats: colsum + colsumsq
  float* ss    = ws + 256;  // 256 floats: scale + shift
  float* Wt1   = ws + 512;                    // 16384 floats
  float* Wt2   = Wt1 + D_DIM * D_DIM;         // 16384 floats
  float* h     = Wt2 + D_DIM * D_DIM;         // n*128
  float* agg   = h + (size_t)n * D_DIM;       // n*128 (also GEMM1 output)

  zero_stats_kernel<<<1, 256, 0, stream>>>(stats);
  bn_partial_kernel<<<512, D_DIM, 0, stream>>>(x, stats, n);
  bn_finalize_kernel<<<1, D_DIM, 0, stream>>>(stats, gamma, beta, ss, n);

  // transpose both weight matrices once (tiny: 64 KB each, L2 resident)
  transpose_w_kernel<<<(D_DIM * D_DIM) / 256, 256, 0, stream>>>(W1, Wt1);
  transpose_w_kernel<<<(D_DIM * D_DIM) / 256, 256, 0, stream>>>(W2, Wt2);

  int tot4 = n * (D_DIM / 4);
  bn_apply_kernel<<<(tot4 + 255) / 256, 256, 0, stream>>>(x, ss, epsp, h, agg, n);

  long long ethreads = (long long)ne * 4;
  edge_scatter_kernel<<<(int)((ethreads + 255) / 256), 256, 0, stream>>>(
      h, ei, ea, We1, be1, We2, be2, agg, ne);

  int mtiles = (n + 15) / 16;
  // GEMM1: agg <- LeakyReLU(agg @ W1 + b1)   (in-place; tile staged to LDS first)
  node_mlp_gemm_kernel<<<mtiles, 256, 0, stream>>>(agg, Wt1, b1, nullptr, agg, n, 1);
  // GEMM2: out <- (t1 @ W2 + b2) + x
  node_mlp_gemm_kernel<<<mtiles, 256, 0, stream>>>(agg, Wt2, b2, x, (float*)d_out, n, 0);
}